// CardClassifier_9740985828117
// MI455X (gfx1250) — compile-verified
//
#include <hip/hip_runtime.h>
#include <hip/hip_bf16.h>
#include <stdint.h>

typedef __attribute__((ext_vector_type(2))) float v2f;
typedef __attribute__((ext_vector_type(8))) float v8f;
typedef __attribute__((ext_vector_type(4))) unsigned int v4u;
typedef __attribute__((ext_vector_type(8))) int v8i;
typedef __attribute__((ext_vector_type(4))) int v4i;

// ---------------------------------------------------------------------------
// TDM: 2D tile load Global -> LDS (clang-23 6-arg builtin).
// Descriptor per CDNA5 ISA ch.8: group0 = {flags, lds_addr, global_addr, type},
// group1 = {mask/data_size, dims, tile dims, strides}. tensor_dim taken
// relative to the tile origin so out-of-range columns/rows zero-fill LDS.
// ---------------------------------------------------------------------------
__device__ __forceinline__ void tdm_load_2d(unsigned lds_addr, const float* gptr,
                                            unsigned valid_w, unsigned valid_h,
                                            unsigned tile_w, unsigned tile_h,
                                            unsigned row_stride_elems) {
  unsigned long long ga = (unsigned long long)(uintptr_t)gptr;
  v4u g0;
  g0[0] = 1u;                                           // count=1, user mode
  g0[1] = lds_addr;                                     // LDS byte address
  g0[2] = (unsigned)(ga & 0xffffffffu);                 // global addr lo
  g0[3] = (unsigned)((ga >> 32) & 0x1ffffffu) | (2u << 30); // addr[56:32] | type=2
  v8i g1;
  g1[0] = (int)(2u << 16);                              // data_size=4B, no multicast
  g1[1] = (int)((valid_w & 0xffffu) << 16);             // tensor_dim0[15:0]
  g1[2] = (int)((valid_w >> 16) | ((valid_h & 0xffffu) << 16)); // dim0 hi | dim1 lo
  g1[3] = (int)((valid_h >> 16) | (tile_w << 16));      // dim1 hi | tile_dim0
  g1[4] = (int)(tile_h & 0xffffu);                      // tile_dim1 (tile_dim2=0)
  g1[5] = (int)row_stride_elems;                        // tensor_dim0_stride lo32
  g1[6] = 0;
  g1[7] = 0;
  v4i z4 = {0, 0, 0, 0};
  v8i z8 = {0, 0, 0, 0, 0, 0, 0, 0};
  __builtin_amdgcn_tensor_load_to_lds(g0, g1, z4, z4, z8, 0);
}

// ---------------------------------------------------------------------------
// Stage-1 conv (Ci=3, not matrix-friendly): fused direct fp32 conv3x3(SAME)
// + bias + relu + maxpool2x2. relu(max(conv)+b) == max(relu(conv+b)).
// ---------------------------------------------------------------------------
__global__ __launch_bounds__(256) void conv3x3_relu_pool(
    const float* __restrict__ in, const float* __restrict__ w,
    const float* __restrict__ bias, float* __restrict__ out,
    int Ci, int H, int Wd) {
  int OH = H >> 1, OW = Wd >> 1;
  int co = blockIdx.y;
  int b  = blockIdx.z;
  int Co = gridDim.y;
  int p = blockIdx.x * blockDim.x + threadIdx.x;
  if (p >= OH * OW) return;
  int oy = p / OW, ox = p - oy * OW;

  float acc00 = 0.f, acc01 = 0.f, acc10 = 0.f, acc11 = 0.f;
  const float* wc = w + (long long)co * Ci * 9;
  for (int ci = 0; ci < Ci; ++ci) {
    const float* ip = in + ((long long)b * Ci + ci) * H * Wd;
    const float* wp = wc + ci * 9;
    float wk[9];
#pragma unroll
    for (int i = 0; i < 9; ++i) wk[i] = wp[i];
#pragma unroll
    for (int dy = 0; dy < 2; ++dy) {
#pragma unroll
      for (int dx = 0; dx < 2; ++dx) {
        int y = 2 * oy + dy, x = 2 * ox + dx;
        float s = 0.f;
#pragma unroll
        for (int ky = -1; ky <= 1; ++ky) {
          int yy = y + ky;
          if (yy < 0 || yy >= H) continue;
#pragma unroll
          for (int kx = -1; kx <= 1; ++kx) {
            int xx = x + kx;
            if (xx < 0 || xx >= Wd) continue;
            s = fmaf(ip[yy * Wd + xx], wk[(ky + 1) * 3 + (kx + 1)], s);
          }
        }
        if (dy == 0 && dx == 0) acc00 += s;
        else if (dy == 0)       acc01 += s;
        else if (dx == 0)       acc10 += s;
        else                    acc11 += s;
      }
    }
  }
  float m = fmaxf(fmaxf(acc00, acc01), fmaxf(acc10, acc11));
  out[((long long)b * Co + co) * OH * OW + p] = fmaxf(m + bias[co], 0.f);
}

// ---------------------------------------------------------------------------
// WMMA conv3x3(SAME)+bias+relu+maxpool2x2 via 9-shifted-GEMM decomposition:
// each tap (ky,kx) is a 1x1 conv = GEMM with K=16-channel blocks, all taps
// accumulated into the same f32 WMMA accumulators. Block = 256 threads,
// 8 waves = (Co/16 co-tiles) x (SC col strips), covering 2 conv rows x
// SC*16 conv cols x all Co. Input patch (16 ch x 4 rows x SC*16+2 cols,
// zero-padded halo) staged in LDS once per channel block; two accumulators
// (conv rows y, y+1) give two independent WMMA chains. 2x2 maxpool + bias +
// relu fold into the register epilogue: pool pairs are adjacent accumulator
// VGPRs of the same lane, pool rows are the two accumulators.
// Requires Ci % 16 == 0, Co in {64,128}, even H/W.
// ---------------------------------------------------------------------------
__global__ __launch_bounds__(256) void conv3x3_wmma_pool(
    const float* __restrict__ in, const float* __restrict__ w,
    const float* __restrict__ bias, float* __restrict__ out,
    int Ci, int Co, int H, int Wd) {
  __shared__ float P[16][4][34];           // [ci'][patch row][patch col]
  const int NT = Co >> 4;                  // co tiles per block (4 or 8)
  const int SC = 8 / NT;                   // col strips per block (2 or 1)
  const int b  = blockIdx.z;
  const int py = blockIdx.y;               // pooled output row
  const int x0 = blockIdx.x * (SC << 4);   // first conv col of block
  const int OW = Wd >> 1, OH = H >> 1;

  const int tid  = threadIdx.x;
  const int wave = tid >> 5;
  const int lane = tid & 31;
  const int nt    = wave % NT;
  const int strip = wave / NT;
  const int r = lane & 15;
  const int khalf = (lane >> 4) << 1;
  const int co0 = nt << 4;
  const int xs  = strip << 4;              // strip col offset within block
  const int y0  = py * 2;                  // first conv row
  const int cw  = (SC << 4) + 2;           // patch cols in use (18 or 34)

  const int cstage = tid >> 4;             // staging: channel per 16 threads
  const int s0     = tid & 15;             // staging: spatial start

  v8f acc0 = {}, acc1 = {};

  const int nkb = Ci >> 4;
  for (int cib = 0; cib < nkb; ++cib) {
    __syncthreads();
    // ---- stage zero-padded input patch for this 16-channel block ----
    {
      const float* ipc = in + ((long long)b * Ci + cib * 16 + cstage) * H * Wd;
      for (int s = s0; s < 4 * cw; s += 16) {
        int pr = s / cw, pc = s - pr * cw;
        int yy = y0 - 1 + pr, xx = x0 - 1 + pc;
        float v = 0.f;
        if (yy >= 0 && yy < H && xx >= 0 && xx < Wd) v = ipc[yy * Wd + xx];
        P[cstage][pr][pc] = v;
      }
    }
    __syncthreads();
    // ---- 9 taps x 4 K-steps, both conv rows ----
#pragma unroll
    for (int ky = 0; ky < 3; ++ky) {
#pragma unroll
      for (int kx = 0; kx < 3; ++kx) {
        const float* wt =
            w + ((long long)(co0 + r) * Ci + cib * 16 + khalf) * 9 + ky * 3 + kx;
#pragma unroll
        for (int kk = 0; kk < 16; kk += 4) {
          v2f bf, a0, a1;
          bf.x = wt[kk * 9];
          bf.y = wt[kk * 9 + 9];
          a0.x = P[kk + khalf][ky][xs + r + kx];
          a0.y = P[kk + khalf + 1][ky][xs + r + kx];
          a1.x = P[kk + khalf][ky + 1][xs + r + kx];
          a1.y = P[kk + khalf + 1][ky + 1][xs + r + kx];
          acc0 = __builtin_amdgcn_wmma_f32_16x16x4_f32(
              false, a0, false, bf, (short)0, acc0, false, false);
          acc1 = __builtin_amdgcn_wmma_f32_16x16x4_f32(
              false, a1, false, bf, (short)0, acc1, false, false);
        }
      }
    }
  }

  // ---- epilogue: 2x2 maxpool + bias + relu + store ----
  const int co = co0 + r;
  const float bv = bias[co];
  const int xbase = x0 + xs + ((lane >> 4) << 3);  // conv col of this half's m=0
  float* op = out + (((long long)b * Co + co) * OH + py) * OW;
#pragma unroll
  for (int j = 0; j < 4; ++j) {
    int xi = xbase + 2 * j;
    if (xi < Wd) {
      float pm = fmaxf(fmaxf(acc0[2 * j], acc0[2 * j + 1]),
                       fmaxf(acc1[2 * j], acc1[2 * j + 1]));
      op[xi >> 1] = fmaxf(pm + bv, 0.f);
    }
  }
}

// ---------------------------------------------------------------------------
// Attention: logit[b,n] = f[b,n,:].wa + ba; softmax over the 128 feature maps;
// h[b,n,:] = f[b,n,:] * att[b,n].   One block per batch row, 128 threads.
// ---------------------------------------------------------------------------
__global__ __launch_bounds__(128) void attention_scale(
    const float* __restrict__ f, const float* __restrict__ wa,
    const float* __restrict__ ba, float* __restrict__ h) {
  __shared__ float logit[128];
  int b = blockIdx.x, n = threadIdx.x;
  const float* fr = f + ((long long)b * 128 + n) * 784;
  float s = 0.f;
  for (int i = 0; i < 784; ++i) s = fmaf(fr[i], wa[i], s);
  logit[n] = s + ba[0];
  __syncthreads();
  float mx = -3.4e38f;
  for (int i = 0; i < 128; ++i) mx = fmaxf(mx, logit[i]);
  float den = 0.f;
  for (int i = 0; i < 128; ++i) den += __expf(logit[i] - mx);
  float att = __expf(logit[n] - mx) / den;
  float* hr = h + ((long long)b * 128 + n) * 784;
  for (int i = 0; i < 784; ++i) hr[i] = fr[i] * att;
}

// ---------------------------------------------------------------------------
// Batched skinny GEMM, C[32xN] = relu?(A[32xK] * W[KxN] + bias), fp32 WMMA.
// Block = 256 threads = 8 waves = 2 M-tiles x 4 N-tiles over a 64-col strip.
// Double-buffered pipeline: wave 0 issues the TDM for chunk i+1 into the
// alternate LDS buffer while all waves compute chunk i; one s_wait_tensorcnt
// + barrier per chunk publishes the prefetched tile. 32 KB per TDM transfer
// amortizes HBM latency. A-tile staged with b128 global loads. Full-chunk
// compute path has compile-time trip count + unroll so ds_loads batch ahead
// of the dependent WMMA accumulate chain (partial dscnt waits).
// Requires: every K chunk length to be a multiple of 4 (true for all calls).
// ---------------------------------------------------------------------------
#define KCMAX 128
#define APITCH 132   // 128 + 4: keeps b128 LDS stores 16B-aligned, rows on distinct banks
#define NBLK  64

__global__ __launch_bounds__(256) void gemm32_wmma(
    const float* __restrict__ A, int ldA, long long strideAb,
    const float* __restrict__ W, int ldW, long long strideWb,
    const float* __restrict__ bias, long long strideBb,
    float* __restrict__ C, int ldC, long long strideCb,
    int N, int K, int relu) {
  __shared__ float Alds[2][32][APITCH];
  __shared__ float Wlds[2][KCMAX][NBLK];

  const int batch = blockIdx.y;
  const int n0 = blockIdx.x * NBLK;
  const float* Ab = A + (long long)batch * strideAb;
  const float* Wb = W + (long long)batch * strideWb;
  const float* Bb = bias + (long long)batch * strideBb;
  float* Cb = C + (long long)batch * strideCb;

  const int tid  = threadIdx.x;
  const int wave = tid >> 5;
  const int lane = tid & 31;
  const int mt = wave & 1;        // row tile: rows mt*16 .. mt*16+15
  const int nt = wave >> 1;       // col tile: cols nt*16 .. nt*16+15 inside strip
  const int r = lane & 15;
  const int khalf = (lane >> 4) << 1;   // 0 for lanes 0-15, 2 for lanes 16-31

  const int arow = tid >> 3;      // 0..31, 8 threads per A row
  const int acol0 = (tid & 7) * 4;

  const int nchunks = (K + KCMAX - 1) / KCMAX;
  v8f acc = {};

  // ---- prologue: stage chunk 0 into buffer 0 ----
  {
    int kc0 = K < KCMAX ? K : KCMAX;
    if (wave == 0)
      tdm_load_2d((unsigned)(uintptr_t)&Wlds[0][0][0], Wb + n0,
                  (unsigned)(N - n0), (unsigned)K, NBLK, (unsigned)kc0,
                  (unsigned)ldW);
    for (int kk = acol0; kk + 4 <= kc0; kk += 32)
      *(float4*)&Alds[0][arow][kk] =
          *(const float4*)(Ab + (long long)arow * ldA + kk);
    if (wave == 0) __builtin_amdgcn_s_wait_tensorcnt(0);
    __syncthreads();
  }

  int buf = 0;
  for (int ci = 0; ci < nchunks; ++ci) {
    const int k0 = ci * KCMAX;
    int kc = K - k0;
    if (kc > KCMAX) kc = KCMAX;
    const int nb = buf ^ 1;

    // ---- prefetch chunk ci+1 into the alternate buffer ----
    if (ci + 1 < nchunks) {
      const int k1 = k0 + KCMAX;
      int kc1 = K - k1;
      if (kc1 > KCMAX) kc1 = KCMAX;
      if (wave == 0)
        tdm_load_2d((unsigned)(uintptr_t)&Wlds[nb][0][0],
                    Wb + (long long)k1 * ldW + n0,
                    (unsigned)(N - n0), (unsigned)(K - k1), NBLK, (unsigned)kc1,
                    (unsigned)ldW);
      for (int kk = acol0; kk + 4 <= kc1; kk += 32)
        *(float4*)&Alds[nb][arow][kk] =
            *(const float4*)(Ab + (long long)arow * ldA + k1 + kk);
    }

    // ---- compute chunk ci from the current buffer ----
    const float* Arow = &Alds[buf][mt * 16 + r][khalf];
    const float* Wcol = &Wlds[buf][khalf][nt * 16 + r];
    if (kc == KCMAX) {
#pragma unroll 8
      for (int kk = 0; kk < KCMAX; kk += 4) {
        v2f a, b;
        a.x = Arow[kk];
        a.y = Arow[kk + 1];
        b.x = Wcol[kk * NBLK];
        b.y = Wcol[kk * NBLK + NBLK];
        acc = __builtin_amdgcn_wmma_f32_16x16x4_f32(
            false, a, false, b, (short)0, acc, false, false);
      }
    } else {
#pragma unroll 4
      for (int kk = 0; kk < kc; kk += 4) {
        v2f a, b;
        a.x = Arow[kk];
        a.y = Arow[kk + 1];
        b.x = Wcol[kk * NBLK];
        b.y = Wcol[kk * NBLK + NBLK];
        acc = __builtin_amdgcn_wmma_f32_16x16x4_f32(
            false, a, false, b, (short)0, acc, false, false);
      }
    }

    if (ci + 1 < nchunks) {
      if (wave == 0) __builtin_amdgcn_s_wait_tensorcnt(0);
      __syncthreads();
    }
    buf = nb;
  }

  // ---- epilogue: bias + relu + store ----
  int col = n0 + nt * 16 + r;
  if (col < N) {
    float bv = Bb[col];
#pragma unroll
    for (int v = 0; v < 8; ++v) {
      int row = mt * 16 + v + ((lane >> 4) << 3);
      float x = acc[v] + bv;
      if (relu) x = fmaxf(x, 0.f);
      Cb[(long long)row * ldC + col] = x;
    }
  }
}

// ---------------------------------------------------------------------------
extern "C" void kernel_launch(void* const* d_in, const int* in_sizes, int n_in,
                              void* d_out, int out_size, void* d_ws, size_t ws_size,
                              hipStream_t stream) {
  const float* x   = (const float*)d_in[0];
  const float* cw1 = (const float*)d_in[1];
  const float* cb1 = (const float*)d_in[2];
  const float* cw2 = (const float*)d_in[3];
  const float* cb2 = (const float*)d_in[4];
  const float* cw3 = (const float*)d_in[5];
  const float* cb3 = (const float*)d_in[6];
  const float* wa  = (const float*)d_in[7];
  const float* ba  = (const float*)d_in[8];
  const float* ew1 = (const float*)d_in[9];
  const float* eb1 = (const float*)d_in[10];
  const float* ew2 = (const float*)d_in[11];
  const float* eb2 = (const float*)d_in[12];
  const float* ew3 = (const float*)d_in[13];
  const float* eb3 = (const float*)d_in[14];
  const float* fw1 = (const float*)d_in[15];
  const float* fb1 = (const float*)d_in[16];
  const float* fw2 = (const float*)d_in[17];
  const float* fb2 = (const float*)d_in[18];
  const float* fw3 = (const float*)d_in[19];
  const float* fb3 = (const float*)d_in[20];
  float* out = (float*)d_out;

  // Workspace layout (bytes). f1 region is recycled for h/h1/h2, f2 region
  // for e/g1/g2 once the earlier tensor is fully consumed.
  char* ws = (char*)d_ws;
  float* f1 = (float*)(ws + 0);                      // 32*32*112*112 = 51,380,224 B
  float* f2 = (float*)(ws + 51380224);               // 32*64*56*56   = 25,690,112 B
  float* f3 = (float*)(ws + 77070336);               // 32*128*784    = 12,845,056 B
  float* h  = (float*)(ws + 0);                      // reuse f1 region
  float* h1 = (float*)(ws + 12845056);
  float* h2 = (float*)(ws + 25690112);
  float* e  = (float*)(ws + 51380224);               // reuse f2 region
  float* g1 = (float*)(ws + 52428800);
  float* g2 = (float*)(ws + 53477376);

  dim3 b256(256);
  // CNN feature extractor: 224 -> 112 -> 56 -> 28
  conv3x3_relu_pool<<<dim3(49, 32, 32), b256, 0, stream>>>(x, cw1, cb1, f1, 3, 224, 224);
  conv3x3_wmma_pool<<<dim3(4, 56, 32),  b256, 0, stream>>>(f1, cw2, cb2, f2, 32, 64, 112, 112);
  conv3x3_wmma_pool<<<dim3(4, 28, 32),  b256, 0, stream>>>(f2, cw3, cb3, f3, 64, 128, 56, 56);
  // attention softmax over feature maps + scale
  attention_scale<<<dim3(32), dim3(128), 0, stream>>>(f3, wa, ba, h);
  // 128 stacked experts: 784 -> 784 -> 392 -> 64  (batched over grid.y)
  gemm32_wmma<<<dim3(13, 128), b256, 0, stream>>>(h,  100352, 784LL, ew1, 784, 614656LL,
                                                  eb1, 784LL, h1, 100352, 784LL, 784, 784, 1);
  gemm32_wmma<<<dim3(7, 128),  b256, 0, stream>>>(h1, 100352, 784LL, ew2, 392, 307328LL,
                                                  eb2, 392LL, h2, 50176, 392LL, 392, 784, 1);
  gemm32_wmma<<<dim3(1, 128),  b256, 0, stream>>>(h2, 50176,  392LL, ew3, 64,  25088LL,
                                                  eb3, 64LL,  e,  8192,  64LL,  64,  392, 1);
  // head: 8192 -> 8192 -> 4096 -> 53
  gemm32_wmma<<<dim3(128, 1), b256, 0, stream>>>(e,  8192, 0LL, fw1, 8192, 0LL,
                                                 fb1, 0LL, g1, 8192, 0LL, 8192, 8192, 1);
  gemm32_wmma<<<dim3(64, 1),  b256, 0, stream>>>(g1, 8192, 0LL, fw2, 4096, 0LL,
                                                 fb2, 0LL, g2, 4096, 0LL, 4096, 8192, 1);
  gemm32_wmma<<<dim3(1, 1),   b256, 0, stream>>>(g2, 4096, 0LL, fw3, 53,   0LL,
                                                 fb3, 0LL, out, 53,  0LL, 53,   4096, 0);
}